// TransformerLoss_41858751267323
// MI455X (gfx1250) — compile-verified
//
#include <hip/hip_runtime.h>
#include <hip/hip_bf16.h>

#define B_SZ 4096
#define N_PTS 2048
#define PI_OVER_12 0.26179938779914946f

typedef __attribute__((ext_vector_type(2))) float v2f;
typedef __attribute__((ext_vector_type(8))) float v8f;

__device__ __forceinline__ float huber_elem(float a) {
    // a >= 0; delta = 1:  0.5*q^2 + (a - q), q = min(a, 1)
    float q = fminf(a, 1.0f);
    return 0.5f * q * q + (a - q);
}

__global__ __launch_bounds__(256) void frustum_loss_stage1(
    const float* __restrict__ mask_xyz_mean,    // (B,3)
    const float* __restrict__ point_cloud,      // (B,3,N)
    const float* __restrict__ x_delta,          // (B,3,N)
    const float* __restrict__ center_label,     // (B,3)
    const float* __restrict__ size_residual,    // (B,3)
    const float* __restrict__ heading_residual, // (B,)
    const float* __restrict__ mean_sizes,       // (8,3)
    const int*   __restrict__ size_class,       // (B,)
    const int*   __restrict__ heading_class,    // (B,)
    float* __restrict__ ws)                     // (B,) per-box contribution
{
    const int b    = blockIdx.x;
    const int tid  = threadIdx.x;
    const int lane = tid & 31;
    const int wave = tid >> 5;

    const float*  pc  = point_cloud + (size_t)b * (3 * N_PTS);
    const float4* px4 = (const float4*)(pc);            // row 0: x
    const float4* py4 = (const float4*)(pc + N_PTS);    // row 1: y
    const float4* xd4 = (const float4*)(x_delta + (size_t)b * (3 * N_PTS));

    // ---- bandwidth-bound accumulation: 5 moments + sum of delta^2 ----
    float sx = 0.f, sy = 0.f, sxx = 0.f, syy = 0.f, sxy = 0.f, sdd = 0.f;
    #pragma unroll
    for (int i = tid; i < N_PTS / 4; i += 256) {        // 2 iters
        float4 x = px4[i];
        float4 y = py4[i];
        sx  += (x.x + x.y) + (x.z + x.w);
        sy  += (y.x + y.y) + (y.z + y.w);
        sxx += x.x * x.x + x.y * x.y + x.z * x.z + x.w * x.w;
        syy += y.x * y.x + y.y * y.y + y.z * y.z + y.w * y.w;
        sxy += x.x * y.x + x.y * y.y + x.z * y.z + x.w * y.w;
    }
    #pragma unroll
    for (int i = tid; i < (3 * N_PTS) / 4; i += 256) {  // 6 iters
        float4 d = xd4[i];
        sdd += d.x * d.x + d.y * d.y + d.z * d.z + d.w * d.w;
    }

    // ---- intra-wave (wave32) reduction of the 6 accumulators ----
    float acc[6] = {sx, sy, sxx, syy, sxy, sdd};
    #pragma unroll
    for (int k = 16; k >= 1; k >>= 1) {
        #pragma unroll
        for (int i = 0; i < 6; ++i)
            acc[i] += __shfl_xor(acc[i], k, 32);
    }

    __shared__ float s_part[8][6];                       // [wave][acc]
    if (lane == 0) {
        #pragma unroll
        for (int i = 0; i < 6; ++i) s_part[wave][i] = acc[i];
    }
    __syncthreads();

    // ---- cross-wave reduction via V_WMMA_F32_16X16X4_F32 ----
    // A (16x4, f32) packed per ISA layout: lanes 0-15 hold K=0 (vgpr a0) and
    // K=1 (vgpr a1) of row M=lane; lanes 16-31 hold K=2/K=3 of row M=lane-16.
    // Row i sums waves 0..3 of acc i; row i+8 sums waves 4..7 of acc i.
    // B = all-ones(4x16)  =>  D[m][n] = rowsum(m) at full f32 precision.
    float a0 = 0.0f, a1 = 0.0f;
    if (lane < 6)                        { a0 = s_part[0][lane];      a1 = s_part[1][lane];      }
    else if (lane >= 8  && lane < 14)    { a0 = s_part[4][lane - 8];  a1 = s_part[5][lane - 8];  }
    else if (lane >= 16 && lane < 22)    { a0 = s_part[2][lane - 16]; a1 = s_part[3][lane - 16]; }
    else if (lane >= 24 && lane < 30)    { a0 = s_part[6][lane - 24]; a1 = s_part[7][lane - 24]; }

    v2f A;     A[0] = a0;   A[1] = a1;
    v2f Bones; Bones[0] = 1.0f; Bones[1] = 1.0f;
    v8f C = {};
    // 8 args: (neg_a, A, neg_b, B, c_mod, C, reuse_a, reuse_b)
    v8f D = __builtin_amdgcn_wmma_f32_16x16x4_f32(false, A, false, Bones,
                                                  (short)0, C, false, false);

    // D vgpr r: lanes 0-15 -> rowsum(r); lanes 16-31 -> rowsum(r+8).
    float tot[6];
    #pragma unroll
    for (int i = 0; i < 6; ++i) {
        float hi = __shfl(D[i], 16, 32);   // rowsum(i+8) = waves 4..7 of acc i
        tot[i] = D[i] + hi;                // valid on lanes 0..15
    }

    // ---- per-box epilogue (thread 0 only) ----
    if (tid == 0) {
        const float cx0 = center_label[b * 3 + 0];
        const float cy0 = center_label[b * 3 + 1];
        const float cz0 = center_label[b * 3 + 2];

        float dx = mask_xyz_mean[b * 3 + 0] - cx0;
        float dy = mask_xyz_mean[b * 3 + 1] - cy0;
        float dz = mask_xyz_mean[b * 3 + 2] - cz0;
        float center_dist = sqrtf(dx * dx + dy * dy + dz * dz);

        int   sc = size_class[b];
        float l  = mean_sizes[sc * 3 + 0] + size_residual[b * 3 + 0];
        float w  = mean_sizes[sc * 3 + 1] + size_residual[b * 3 + 1];
        float heading = heading_residual[b] + (float)heading_class[b] * PI_OVER_12;
        float ch = cosf(heading), sh = sinf(heading);
        float hx = 0.5f * l, hy = 0.5f * w;

        float r    = sqrtf(cx0 * cx0 + cy0 * cy0);
        float invr = 1.0f / r;

        // corners j = 4..7 in xy-plane; project onto p_dir=(-cy,cx)/r and u=(cx,cy)/r
        const float sxs[4] = {1.f, 1.f, -1.f, -1.f};
        const float sys[4] = {1.f, -1.f, -1.f, 1.f};
        float ppmin =  1e30f, ppmax = -1e30f;
        float prmin =  1e30f, prmax = -1e30f;
        #pragma unroll
        for (int j = 0; j < 4; ++j) {
            float px = ch * hx * sxs[j] - sh * hy * sys[j] + cx0;
            float py = sh * hx * sxs[j] + ch * hy * sys[j] + cy0;
            float pp = (-cy0 * px + cx0 * py) * invr;
            float pr = ( cx0 * px + cy0 * py) * invr;
            ppmin = fminf(ppmin, pp); ppmax = fmaxf(ppmax, pp);
            prmin = fminf(prmin, pr); prmax = fmaxf(prmax, pr);
        }
        float std_y_label  = (ppmax - ppmin) * 0.25f;
        float mean_y_label = (ppmax + ppmin) * 0.5f;
        float std_x_label  = (prmax - prmin) * 0.25f;
        float mean_x_label = (prmax + prmin) * 0.5f;

        float Sx = tot[0], Sy = tot[1], Sxx = tot[2], Syy = tot[3], Sxy = tot[4], Sdd = tot[5];
        const float invN   = 1.0f / (float)N_PTS;
        const float invNm1 = 1.0f / (float)(N_PTS - 1);

        float pd0 = -cy0 * invr, pd1 = cx0 * invr;   // p_dir
        float u0  =  cx0 * invr, u1  = cy0 * invr;   // v / |v|

        float mean_y_pc = (pd0 * Sx + pd1 * Sy) * invN;
        float ss_y  = pd0 * pd0 * Sxx + 2.0f * pd0 * pd1 * Sxy + pd1 * pd1 * Syy;
        float var_y = (ss_y - (float)N_PTS * mean_y_pc * mean_y_pc) * invNm1;
        float std_y_pc = sqrtf(fmaxf(var_y, 0.0f));

        float mean_x_pc = (u0 * Sx + u1 * Sy) * invN;
        float ss_x  = u0 * u0 * Sxx + 2.0f * u0 * u1 * Sxy + u1 * u1 * Syy;
        float var_x = (ss_x - (float)N_PTS * mean_x_pc * mean_x_pc) * invNm1;
        float std_x_pc = sqrtf(fmaxf(var_x, 0.0f));

        float delta_norm = sqrtf(Sdd);

        float contrib = 0.5f  * huber_elem(center_dist)
                      +         huber_elem(fabsf(std_x_label  - std_x_pc))
                      +         huber_elem(fabsf(mean_x_label - mean_x_pc))
                      + 0.01f * huber_elem(delta_norm)
                      +         huber_elem(fabsf(mean_y_label - mean_y_pc))
                      +         huber_elem(fabsf(std_y_label  - std_y_pc));

        ws[b] = contrib * (0.4f / (float)B_SZ);
    }
}

// Deterministic fixed-shape tree sum of B_SZ per-box contributions.
__global__ __launch_bounds__(1024) void frustum_loss_stage2(
    const float* __restrict__ ws, float* __restrict__ out)
{
    __shared__ float sm[1024];
    int t = threadIdx.x;
    sm[t] = (ws[t] + ws[t + 1024]) + (ws[t + 2048] + ws[t + 3072]);
    __syncthreads();
    #pragma unroll
    for (int k = 512; k >= 1; k >>= 1) {
        if (t < k) sm[t] += sm[t + k];
        __syncthreads();
    }
    if (t == 0) out[0] = sm[0];
}

extern "C" void kernel_launch(void* const* d_in, const int* in_sizes, int n_in,
                              void* d_out, int out_size, void* d_ws, size_t ws_size,
                              hipStream_t stream) {
    const float* mask_xyz_mean    = (const float*)d_in[0];
    const float* point_cloud      = (const float*)d_in[1];
    const float* x_delta          = (const float*)d_in[2];
    const float* center_label     = (const float*)d_in[3];
    const float* size_residual    = (const float*)d_in[4];
    const float* heading_residual = (const float*)d_in[5];
    const float* mean_sizes       = (const float*)d_in[6];
    const int*   size_class       = (const int*)d_in[7];
    const int*   heading_class    = (const int*)d_in[8];

    float* ws  = (float*)d_ws;     // needs B_SZ * 4 = 16 KB scratch
    float* out = (float*)d_out;

    frustum_loss_stage1<<<B_SZ, 256, 0, stream>>>(
        mask_xyz_mean, point_cloud, x_delta, center_label, size_residual,
        heading_residual, mean_sizes, size_class, heading_class, ws);
    frustum_loss_stage2<<<1, 1024, 0, stream>>>(ws, out);
}